// MGIN_53558242181729
// MI455X (gfx1250) — compile-verified
//
#include <hip/hip_runtime.h>
#include <hip/hip_bf16.h>

#define NN 512
#define EE 16384
#define DD 1280
#define LMD 1024
#define NFD 256
#define MM 64

typedef __attribute__((ext_vector_type(2))) float v2f;
typedef __attribute__((ext_vector_type(8))) float v8f;

// Monotone mapping float -> uint so unsigned atomicMax == IEEE float max.
// key(x) for any non-NaN float is > 0, so key==0 is a safe "no edge" sentinel.
__device__ __forceinline__ unsigned fkey(float x) {
    unsigned b = __float_as_uint(x);
    return (b & 0x80000000u) ? ~b : (b | 0x80000000u);
}
__device__ __forceinline__ float funkey(unsigned k) {
    return __uint_as_float((k & 0x80000000u) ? (k & 0x7FFFFFFFu) : ~k);
}

// 1) h = concat(lm[0,1:-1,:], node_feat); key = 0
__global__ void k_build_h(const float* __restrict__ lm, const float* __restrict__ nf,
                          float* __restrict__ h, unsigned* __restrict__ key) {
    int i = blockIdx.x * blockDim.x + threadIdx.x;
    if (i >= NN * DD) return;
    int n = i / DD, d = i - n * DD;
    float v = (d < LMD) ? lm[(size_t)(n + 1) * LMD + d] : nf[(size_t)n * NFD + (d - LMD)];
    h[i] = v;
    key[i] = 0u;
}

// 2) segment-max scatter: one block per edge, atomic max over ordered keys
__global__ void k_scatter(const float* __restrict__ h, const float* __restrict__ ef,
                          const int* __restrict__ src, const int* __restrict__ dst,
                          unsigned* __restrict__ key) {
    int e = blockIdx.x;
    int s = src[e], t = dst[e];
    float w = ef[e];
    const float* hs = h + (size_t)s * DD;
    unsigned* kd = key + (size_t)t * DD;
    for (int d = threadIdx.x; d < DD; d += blockDim.x) {
        atomicMax(&kd[d], fkey(hs[d] * w));
    }
}

// 3) x = h + neigh  (written in place over key buffer)
__global__ void k_decode(const float* __restrict__ h, unsigned* __restrict__ keyx) {
    int i = blockIdx.x * blockDim.x + threadIdx.x;
    if (i >= NN * DD) return;
    unsigned k = keyx[i];
    float neigh = (k == 0u) ? 0.0f : funkey(k);
    ((float*)keyx)[i] = h[i] + neigh;
}

// 4) node_output = x @ W_g + b_g + h   via V_WMMA_F32_16X16X4_F32
//    Block: 256 threads = 8 waves; block tile = 16 rows x 128 cols; wave tile 16x16.
__global__ void __launch_bounds__(256) k_gemm(const float* __restrict__ x,
                                              const float* __restrict__ Wg,
                                              const float* __restrict__ bg,
                                              const float* __restrict__ h,
                                              float* __restrict__ no) {
    __shared__ float xs[16][132];                 // 16x128 A chunk, padded (stride 132 -> bank step 4)
    const int m0   = blockIdx.y * 16;
    const int n0   = blockIdx.x * 128 + (threadIdx.x >> 5) * 16;
    const int lane = threadIdx.x & 31;
    const int half = lane >> 4;                   // 0: K pair {0,1}, 1: K pair {2,3}
    const int lr   = lane & 15;

    v8f acc = {0.f, 0.f, 0.f, 0.f, 0.f, 0.f, 0.f, 0.f};

    for (int kc = 0; kc < DD; kc += 128) {
        // cooperative, coalesced stage of A chunk (shared by all 8 waves)
        for (int t = threadIdx.x; t < 16 * 128; t += 256) {
            int r = t >> 7, c = t & 127;
            xs[r][c] = x[(size_t)(m0 + r) * DD + kc + c];
        }
        __syncthreads();

        for (int ks = 0; ks < 128; ks += 4) {
            int k0 = ks + 2 * half;
            // A 16x4 frag: lane holds row lr, K = k0 (v0) and k0+1 (v1)
            v2f a = { xs[lr][k0], xs[lr][k0 + 1] };
            // B 4x16 frag: lane holds col n0+lr, rows k0 / k0+1 (coalesced across lanes)
            const float* wb = Wg + (size_t)(kc + k0) * DD + n0 + lr;
            v2f b = { wb[0], wb[DD] };
            acc = __builtin_amdgcn_wmma_f32_16x16x4_f32(
                false, a, false, b, (short)0, acc, false, false);
        }
        __syncthreads();
    }

    // C/D layout: VGPR r = row (half*8 + r), col = n0 + lr. Fuse bias + residual.
    for (int r = 0; r < 8; ++r) {
        int row = m0 + half * 8 + r;
        int col = n0 + lr;
        size_t idx = (size_t)row * DD + col;
        no[idx] = acc[r] + bg[col] + h[idx];
    }
}

// 5) p[n] = node_output[n] . W_d   (one wave per node)
__global__ void k_p(const float* __restrict__ no, const float* __restrict__ Wd,
                    float* __restrict__ p) {
    int node = blockIdx.x * 8 + (threadIdx.x >> 5);
    int lane = threadIdx.x & 31;
    const float* row = no + (size_t)node * DD;
    float acc = 0.f;
    for (int d = lane; d < DD; d += 32) acc += row[d] * Wd[d];
    for (int off = 16; off > 0; off >>= 1) acc += __shfl_xor(acc, off, 32);
    if (lane == 0) p[node] = acc;
}

// 6) dis_pred[i,j] = sigmoid(p[j] - p[i] + b_d)
__global__ void k_dis(const float* __restrict__ p, const float* __restrict__ bd,
                      float* __restrict__ out) {
    int idx = blockIdx.x * blockDim.x + threadIdx.x;
    if (idx >= NN * NN) return;
    int i = idx >> 9, j = idx & (NN - 1);
    float z = p[j] - p[i] + bd[0];
    out[idx] = 1.0f / (1.0f + expf(-z));
}

// 7) mask_pred = tanh(node_output[mask_index] @ W_m + b_m)  (one wave per masked node)
__global__ void k_mask(const float* __restrict__ no, const int* __restrict__ mi,
                       const float* __restrict__ Wm, const float* __restrict__ bm,
                       float* __restrict__ out) {
    int m = blockIdx.x * 8 + (threadIdx.x >> 5);
    int lane = threadIdx.x & 31;
    int row = mi[m];
    const float* r = no + (size_t)row * DD;
    float a0 = 0.f, a1 = 0.f;
    for (int d = lane; d < DD; d += 32) {
        float v = r[d];
        a0 += v * Wm[d * 2 + 0];
        a1 += v * Wm[d * 2 + 1];
    }
    for (int off = 16; off > 0; off >>= 1) {
        a0 += __shfl_xor(a0, off, 32);
        a1 += __shfl_xor(a1, off, 32);
    }
    if (lane == 0) {
        out[m * 2 + 0] = tanhf(a0 + bm[0]);
        out[m * 2 + 1] = tanhf(a1 + bm[1]);
    }
}

extern "C" void kernel_launch(void* const* d_in, const int* in_sizes, int n_in,
                              void* d_out, int out_size, void* d_ws, size_t ws_size,
                              hipStream_t stream) {
    (void)in_sizes; (void)n_in; (void)out_size; (void)ws_size;
    const float* lm  = (const float*)d_in[0];
    const float* nf  = (const float*)d_in[1];
    const float* ef  = (const float*)d_in[2];
    const int*   src = (const int*)d_in[3];
    const int*   dst = (const int*)d_in[4];
    const int*   mi  = (const int*)d_in[5];
    const float* Wg  = (const float*)d_in[6];
    const float* bg  = (const float*)d_in[7];
    const float* Wd  = (const float*)d_in[8];
    const float* bd  = (const float*)d_in[9];
    const float* Wm  = (const float*)d_in[10];
    const float* bm  = (const float*)d_in[11];
    float* out = (float*)d_out;

    // workspace layout (fully initialized every call)
    float*    h   = (float*)d_ws;                        // N*D
    unsigned* key = (unsigned*)(h + (size_t)NN * DD);    // N*D, reused as x (float)
    float*    x   = (float*)key;
    float*    no  = x + (size_t)NN * DD;                 // N*D
    float*    p   = no + (size_t)NN * DD;                // N

    const int nd = NN * DD;
    k_build_h<<<(nd + 255) / 256, 256, 0, stream>>>(lm, nf, h, key);
    k_scatter<<<EE, 256, 0, stream>>>(h, ef, src, dst, key);
    k_decode<<<(nd + 255) / 256, 256, 0, stream>>>(h, key);
    k_gemm<<<dim3(DD / 128, NN / 16), 256, 0, stream>>>(x, Wg, bg, h, no);
    k_p<<<NN / 8, 256, 0, stream>>>(no, Wd, p);
    k_dis<<<(NN * NN + 255) / 256, 256, 0, stream>>>(p, bd, out);
    k_mask<<<MM / 8, 256, 0, stream>>>(no, mi, Wm, bm, out + (size_t)NN * NN);
}